// Weighted_least_squares_80436147519856
// MI455X (gfx1250) — compile-verified
//
#include <hip/hip_runtime.h>
#include <stdint.h>

// Problem constants (from reference)
#define N_      32
#define C_      4
#define HW_     131072      // H * WIDTH = 256*512
#define CHUNKS_ 64          // HW slices per n
#define PTS_    (HW_ / CHUNKS_)   // 2048 points per workgroup
#define TPB_    256         // 8 wave32 per workgroup

typedef uint32_t u32x4 __attribute__((ext_vector_type(4)));
typedef int      i32x8 __attribute__((ext_vector_type(8)));
typedef int      i32x4 __attribute__((ext_vector_type(4)));

// AS3 -> generic cast puts the LDS byte offset in the low 32 bits of the flat
// address (aperture lives in the high half) — exactly what D#.lds_addr wants.
__device__ __forceinline__ uint32_t lds_offset(const void* p) {
    return (uint32_t)(uintptr_t)p;
}

// Issue one TDM 2D tile load: copies a tile from global memory at gaddr (rows stride0_elems apart) to LDS at lds_addr
// D# packing per CDNA5 ISA §8.3 (group0) / §8.4 (group1); groups 2/3 zero (2D tile).
__device__ __forceinline__ void tdm_load_tile(
    uint32_t lds_addr, uint64_t gaddr,
    uint32_t tensor_d0, uint32_t tensor_d1,
    uint32_t tile_d0,   uint32_t tile_d1,
    uint64_t stride0_elems)
{
    u32x4 g0;
    g0[0] = 1u;                                   // count=1 (valid user descriptor)
    g0[1] = lds_addr;                             // lds_addr [63:32]
    g0[2] = (uint32_t)gaddr;                      // global_addr lo
    g0[3] = (uint32_t)((gaddr >> 32) & 0x01FFFFFFu) | (2u << 30); // addr[56:32] | type=2

    // group1, built as 4 little-endian qwords (bit numbering == dword order)
    // q0: [17:16]=data_size(2 => 4B), [63:48]=tensor_dim0[15:0]
    uint64_t q0 = (2ull << 16) | ((uint64_t)(tensor_d0 & 0xFFFFu) << 48);
    // q1: [15:0]=tensor_dim0[31:16], [47:16]=tensor_dim1, [63:48]=tile_dim0
    uint64_t q1 = (uint64_t)(tensor_d0 >> 16)
                | ((uint64_t)tensor_d1 << 16)
                | ((uint64_t)(tile_d0 & 0xFFFFu) << 48);
    // q2: [15:0]=tile_dim1, [31:16]=tile_dim2(0), [63:32]=stride0[31:0]
    uint64_t q2 = (uint64_t)(tile_d1 & 0xFFFFu)
                | ((stride0_elems & 0xFFFFFFFFull) << 32);
    // q3: [15:0]=stride0[47:32], [63:16]=tensor_dim1_stride (unused for 2D tile)
    uint64_t q3 = (stride0_elems >> 32) & 0xFFFFull;

    i32x8 g1;
    g1[0] = (int)(uint32_t)(q0      ); g1[1] = (int)(uint32_t)(q0 >> 32);
    g1[2] = (int)(uint32_t)(q1      ); g1[3] = (int)(uint32_t)(q1 >> 32);
    g1[4] = (int)(uint32_t)(q2      ); g1[5] = (int)(uint32_t)(q2 >> 32);
    g1[6] = (int)(uint32_t)(q3      ); g1[7] = (int)(uint32_t)(q3 >> 32);

    i32x4 gz4 = {0, 0, 0, 0};
    i32x8 gz8 = {0, 0, 0, 0, 0, 0, 0, 0};
    // clang-23 / therock-10.0 form: (g0, g1, g2, g3, <extra group>, cpol)
    __builtin_amdgcn_tensor_load_to_lds(g0, g1, gz4, gz4, gz8, 0);
}

// ---------------------------------------------------------------------------
// Kernel 1: per-(n, chunk) partial moments.
// Workgroup = one n and one 2048-point slice. TDM streams W (4 class rows) and
// grid (x,y interleaved) into LDS; 256 threads accumulate the 8 moments per
// class; hierarchical reduce (wave shuffle -> LDS -> 32 lanes) writes 32
// partial sums per workgroup to the workspace.
// ---------------------------------------------------------------------------
__global__ __launch_bounds__(TPB_) void wls_partial(
    const float* __restrict__ W, const float* __restrict__ G,
    float* __restrict__ ws)
{
    __shared__ float sW[C_ * PTS_];     // 32 KiB: W[n, c, pbase..pbase+2047]
    __shared__ float sG[2 * PTS_];      // 16 KiB: grid[n, pbase.., 0:2]
    __shared__ float sRed[8 * 32];      // per-wave partials

    const int n     = blockIdx.x / CHUNKS_;
    const int chunk = blockIdx.x % CHUNKS_;
    const int pbase = chunk * PTS_;
    const int tid   = threadIdx.x;

    if (tid < 32) {   // wave 0 issues both TDM descriptors
        uint64_t wAddr = (uint64_t)(uintptr_t)(W + ((size_t)n * C_ * HW_ + pbase));
        tdm_load_tile(lds_offset(sW), wAddr,
                      (uint32_t)HW_, (uint32_t)C_,     // tensor dims
                      (uint32_t)PTS_, (uint32_t)C_,    // tile: 2048 x 4
                      (uint64_t)HW_);                  // class-row stride
        uint64_t gAddr = (uint64_t)(uintptr_t)(G + ((size_t)n * HW_ + pbase) * 2);
        tdm_load_tile(lds_offset(sG), gAddr,
                      (uint32_t)(2 * PTS_), 1u,
                      (uint32_t)(2 * PTS_), 1u,        // tile: 4096 x 1 (1D)
                      (uint64_t)(2 * PTS_));
        __builtin_amdgcn_s_wait_tensorcnt(0);
    }
    __syncthreads();

    // acc[c*8 + k]: k=0..4 -> m_k = sum w^2 y^k ; k=5..7 -> r_{k-5} = sum w^2 x y^(k-5)
    float acc[32];
#pragma unroll
    for (int j = 0; j < 32; ++j) acc[j] = 0.0f;

#pragma unroll
    for (int k = 0; k < PTS_ / TPB_; ++k) {           // 8 iterations
        const int p = k * TPB_ + tid;                 // conflict-free LDS access
        const float x  = sG[2 * p + 0];
        const float y  = 1.0f - sG[2 * p + 1];
        const float y2 = y * y, y3 = y2 * y, y4 = y2 * y2;
#pragma unroll
        for (int c = 0; c < C_; ++c) {
            const float w  = sW[c * PTS_ + p];
            const float w2 = w * w;
            const float wx = w2 * x;
            float* a = acc + c * 8;
            a[0] += w2;      a[1] += w2 * y; a[2] += w2 * y2;
            a[3] += w2 * y3; a[4] += w2 * y4;
            a[5] += wx;      a[6] += wx * y; a[7] += wx * y2;
        }
    }

    // Intra-wave tree reduction (wave32)
#pragma unroll
    for (int j = 0; j < 32; ++j) {
        float v = acc[j];
        for (int o = 16; o > 0; o >>= 1) v += __shfl_down(v, o, 32);
        acc[j] = v;
    }
    const int lane = tid & 31, wave = tid >> 5;
    if (lane == 0) {
#pragma unroll
        for (int j = 0; j < 32; ++j) sRed[wave * 32 + j] = acc[j];
    }
    __syncthreads();
    if (tid < 32) {
        float s = 0.0f;
#pragma unroll
        for (int w8 = 0; w8 < 8; ++w8) s += sRed[w8 * 32 + tid];
        ws[(size_t)blockIdx.x * 32 + tid] = s;
    }
}

// ---------------------------------------------------------------------------
// Kernel 2: fold 64 chunk-partials per (n,c), build Z = moments + I, solve the
// 3x3 system by adjugate (Cramer), write beta in the reference's return order:
// out[c*N*3 + n*3 + i].
// ---------------------------------------------------------------------------
__global__ __launch_bounds__(128) void wls_solve(
    const float* __restrict__ ws, float* __restrict__ out)
{
    const int t = threadIdx.x;
    if (t >= 128) return;
    const int c = t >> 5;      // class
    const int n = t & 31;      // batch

    float S[8];
#pragma unroll
    for (int j = 0; j < 8; ++j) S[j] = 0.0f;
    for (int ch = 0; ch < CHUNKS_; ++ch) {
        const float* p = ws + ((size_t)(n * CHUNKS_ + ch) * 32 + c * 8);
#pragma unroll
        for (int j = 0; j < 8; ++j) S[j] += p[j];
    }
    const float m0 = S[0], m1 = S[1], m2 = S[2], m3 = S[3], m4 = S[4];
    const float r0 = S[5], r1 = S[6], r2 = S[7];

    // Z + REG*I, REG = 1.0
    const float a = m4 + 1.0f, b = m3,        cc = m2;
    const float d = m3,        e = m2 + 1.0f, f  = m1;
    const float g = m2,        h = m1,        i  = m0 + 1.0f;

    const float A =  (e * i - f * h), B = -(d * i - f * g), Cf =  (d * h - e * g);
    const float D = -(b * i - cc * h), E =  (a * i - cc * g), F = -(a * h - b * g);
    const float Gc =  (b * f - cc * e), H = -(a * f - cc * d), I =  (a * e - b * d);
    const float inv = 1.0f / (a * A + b * B + cc * Cf);

    // rhs = [r2, r1, r0];  beta = adj(Z)^T * rhs / det
    const float x0 = r2, x1 = r1, x2 = r0;
    const float b0 = (A  * x0 + D * x1 + Gc * x2) * inv;
    const float b1 = (B  * x0 + E * x1 + H  * x2) * inv;
    const float b2 = (Cf * x0 + F * x1 + I  * x2) * inv;

    float* o = out + ((size_t)c * N_ + n) * 3;
    o[0] = b0; o[1] = b1; o[2] = b2;
}

extern "C" void kernel_launch(void* const* d_in, const int* in_sizes, int n_in,
                              void* d_out, int out_size, void* d_ws, size_t ws_size,
                              hipStream_t stream) {
    (void)in_sizes; (void)n_in; (void)out_size; (void)ws_size;
    const float* W = (const float*)d_in[0];   // (N, C, H, W) f32
    const float* G = (const float*)d_in[1];   // (N, HW, 2)   f32
    float* ws  = (float*)d_ws;                // 2048 * 32 floats = 256 KiB
    float* out = (float*)d_out;               // 384 floats

    wls_partial<<<N_ * CHUNKS_, TPB_, 0, stream>>>(W, G, ws);
    wls_solve<<<1, 128, 0, stream>>>(ws, out);
}